// LengthRegulator_39084202394283
// MI455X (gfx1250) — compile-verified
//
#include <hip/hip_runtime.h>
#include <hip/hip_bf16.h>

// LengthRegulator for MI455X (gfx1250, wave32).
// B=32, S=2048, d=384 (= 96 x 16B = 3 x b128 per lane). T derived from out_size.
//
// Roofline: 0 FLOPs, ~100MB read + ~350MB write -> HBM-bound (~20us @ 23.3TB/s).
// WMMA is intentionally not used (would only add traffic). The gfx1250 paths
// exercised instead: async global<->LDS B128 DMA (ASYNCcnt), s_wait_asynccnt,
// non-temporal B128 stores, wave32-shaped row tiling.

typedef float f32x4 __attribute__((ext_vector_type(4)));

#define LR_B   32
#define LR_S   2048
#define LR_D4  96          // d in 16-byte units (384 floats)
#define SCAN_BLK 256
#define SCAN_PER (LR_S / SCAN_BLK)   // 8 elements per thread

#if defined(__HIP_DEVICE_COMPILE__) &&                                   \
    __has_builtin(__builtin_amdgcn_global_load_async_to_lds_b128) &&     \
    __has_builtin(__builtin_amdgcn_global_store_async_from_lds_b128)
#define HAVE_ASYNC_LDS 1
// The builtins take pointers to a 16-byte int vector (per hipcc diagnostic):
//   arg0: global (AS1 / __device__) v4i*,  arg1: LDS (AS3 / __shared__) v4i*
typedef int lr_v4i __attribute__((vector_size(4 * sizeof(int))));
typedef __attribute__((address_space(1))) lr_v4i as1_v4i;
typedef __attribute__((address_space(3))) lr_v4i as3_v4i;
#else
#define HAVE_ASYNC_LDS 0
#endif

static __device__ __forceinline__ void lr_wait_asynccnt0() {
#if HAVE_ASYNC_LDS
#if __has_builtin(__builtin_amdgcn_s_wait_asynccnt)
    __builtin_amdgcn_s_wait_asynccnt(0);
#else
    asm volatile("s_wait_asynccnt 0x0" ::: "memory");
#endif
#endif
}

// ---------------------------------------------------------------------------
// Kernel 1: per-batch-row inclusive cumsum of durations; also emit row totals.
// One block per batch row. 256 threads x 8 serial elements + LDS block scan.
// ---------------------------------------------------------------------------
__global__ void lr_cumsum_kernel(const int* __restrict__ dur,
                                 int* __restrict__ csum,
                                 int* __restrict__ totals) {
    const int b = blockIdx.x;
    const int t = threadIdx.x;
    const int* __restrict__ row  = dur  + (size_t)b * LR_S;
    int*       __restrict__ crow = csum + (size_t)b * LR_S;

    __shared__ int part[SCAN_BLK];

    int v[SCAN_PER];
    const int base = t * SCAN_PER;
    int acc = 0;
#pragma unroll
    for (int i = 0; i < SCAN_PER; ++i) { v[i] = row[base + i]; acc += v[i]; }
    part[t] = acc;
    __syncthreads();

    // Hillis-Steele inclusive scan over the 256 partials in LDS.
#pragma unroll
    for (int off = 1; off < SCAN_BLK; off <<= 1) {
        int x = (t >= off) ? part[t - off] : 0;
        __syncthreads();
        part[t] += x;
        __syncthreads();
    }

    int run = (t == 0) ? 0 : part[t - 1];   // exclusive prefix of this chunk
#pragma unroll
    for (int i = 0; i < SCAN_PER; ++i) { run += v[i]; crow[base + i] = run; }

    if (t == SCAN_BLK - 1) totals[b] = part[SCAN_BLK - 1];
}

// ---------------------------------------------------------------------------
// Kernel 2: scatter-replicate. One wave32 per input element (b,s).
// Async path: DMA the 1536B row global->LDS (3 x b128/lane), wait ASYNCcnt,
// then stream `dur` copies LDS->global without touching VGPRs.
// Fallback: registers + non-temporal B128 loads/stores.
// ---------------------------------------------------------------------------
__global__ void lr_replicate_kernel(const f32x4* __restrict__ emb,
                                    const int* __restrict__ csum,
                                    f32x4* __restrict__ out,
                                    int T) {
    const int lane = threadIdx.x & 31;
    const int wv   = threadIdx.x >> 5;                 // wave within block
    const int w    = blockIdx.x * 8 + wv;              // global wave id = b*S + s
    const int s    = w & (LR_S - 1);
    const int b    = w >> 11;                          // w / 2048

    const int end   = csum[w];
    const int start = (s == 0) ? 0 : csum[w - 1];
    const int dur   = end - start;                     // wave-uniform

#if HAVE_ASYNC_LDS
    __shared__ f32x4 tile[8 * LR_D4];                  // 12KB: one row per wave
    if (dur <= 0) return;                              // whole wave exits together

    const f32x4* gsrc = emb + (size_t)w * LR_D4 + lane;
    f32x4*       lrow = &tile[wv * LR_D4 + lane];
    f32x4*       gdst = out + ((size_t)b * T + start) * LR_D4 + lane;

    // global -> LDS, 16B per lane per op, imm offsets cover the 1536B row
    __builtin_amdgcn_global_load_async_to_lds_b128((as1_v4i*)gsrc, (as3_v4i*)lrow,    0, 0);
    __builtin_amdgcn_global_load_async_to_lds_b128((as1_v4i*)gsrc, (as3_v4i*)lrow,  512, 0);
    __builtin_amdgcn_global_load_async_to_lds_b128((as1_v4i*)gsrc, (as3_v4i*)lrow, 1024, 0);
    lr_wait_asynccnt0();                               // LDS row is ready

    for (int j = 0; j < dur; ++j, gdst += LR_D4) {
        __builtin_amdgcn_global_store_async_from_lds_b128((as1_v4i*)gdst, (as3_v4i*)lrow,    0, 0);
        __builtin_amdgcn_global_store_async_from_lds_b128((as1_v4i*)gdst, (as3_v4i*)lrow,  512, 0);
        __builtin_amdgcn_global_store_async_from_lds_b128((as1_v4i*)gdst, (as3_v4i*)lrow, 1024, 0);
    }
    // s_endpgm performs an implicit wait-idle, draining ASYNCcnt.
#else
    if (dur <= 0) return;

    const f32x4* gsrc = emb + (size_t)w * LR_D4 + lane;
    f32x4*       gdst = out + ((size_t)b * T + start) * LR_D4 + lane;

    __builtin_prefetch((const void*)(gsrc + 8 * LR_D4), 0, 3); // warm next rows
    const f32x4 r0 = __builtin_nontemporal_load(gsrc);
    const f32x4 r1 = __builtin_nontemporal_load(gsrc + 32);
    const f32x4 r2 = __builtin_nontemporal_load(gsrc + 64);
    for (int j = 0; j < dur; ++j, gdst += LR_D4) {
        __builtin_nontemporal_store(r0, gdst);
        __builtin_nontemporal_store(r1, gdst + 32);
        __builtin_nontemporal_store(r2, gdst + 64);
    }
#endif
}

// ---------------------------------------------------------------------------
// Kernel 3: zero-fill the ragged tail rows t in [total[b], T).
// One wave32 per (b, t); waves inside the valid region exit immediately.
// ---------------------------------------------------------------------------
__global__ void lr_zerotail_kernel(f32x4* __restrict__ out,
                                   const int* __restrict__ totals,
                                   int T) {
    const int lane = threadIdx.x & 31;
    const int t    = blockIdx.x * 8 + (threadIdx.x >> 5);
    const int b    = blockIdx.y;
    if (t >= T) return;
    if (t < totals[b]) return;

    const f32x4 z = {0.f, 0.f, 0.f, 0.f};
    f32x4* gdst = out + ((size_t)b * T + t) * LR_D4 + lane;
    __builtin_nontemporal_store(z, gdst);
    __builtin_nontemporal_store(z, gdst + 32);
    __builtin_nontemporal_store(z, gdst + 64);
}

// ---------------------------------------------------------------------------
extern "C" void kernel_launch(void* const* d_in, const int* in_sizes, int n_in,
                              void* d_out, int out_size, void* d_ws, size_t ws_size,
                              hipStream_t stream) {
    const f32x4* emb = (const f32x4*)d_in[0];   // (B, S, d) float32
    const int*   dur = (const int*)d_in[1];     // (B, S)    int32
    f32x4*       out = (f32x4*)d_out;           // (B, T, d) float32

    const int d_elems = 4 * LR_D4;              // 384
    const int T = out_size / (LR_B * d_elems);  // padded output length

    // Workspace: csum (B*S ints) then totals (B ints). 256KB + 128B.
    int* csum   = (int*)d_ws;
    int* totals = csum + LR_B * LR_S;

    // 1) per-row inclusive cumsum of durations
    lr_cumsum_kernel<<<LR_B, SCAN_BLK, 0, stream>>>(dur, csum, totals);

    // 2) scatter-replicate: one wave per (b,s) -> B*S/8 blocks of 256 threads
    const int nWaves = LR_B * LR_S;             // 65536
    lr_replicate_kernel<<<nWaves / 8, 256, 0, stream>>>(emb, csum, out, T);

    // 3) zero the ragged tail
    dim3 zgrid((T + 7) / 8, LR_B);
    lr_zerotail_kernel<<<zgrid, 256, 0, stream>>>(out, totals, T);
}